// MyFPModule_39874476376402
// MI455X (gfx1250) — compile-verified
//
#include <hip/hip_runtime.h>
#include <hip/hip_bf16.h>

// Problem constants from the reference: B=8, N=16384, M=4096, C=256
#define PB 8
#define PN 16384
#define PM 4096
#define PC 256
#define EPSI 1e-8f

typedef __attribute__((ext_vector_type(2))) float v2f;
typedef __attribute__((ext_vector_type(8))) float v8f;

__device__ __forceinline__ void insert3(float val, int idx,
                                        float& d1, float& d2, float& d3,
                                        int& i1, int& i2, int& i3) {
    bool c1 = val < d1, c2 = val < d2, c3 = val < d3;
    float nd3 = c2 ? d2 : val; int ni3 = c2 ? i2 : idx;
    d3 = c3 ? nd3 : d3;        i3 = c3 ? ni3 : i3;
    float nd2 = c1 ? d1 : val; int ni2 = c1 ? i1 : idx;
    d2 = c2 ? nd2 : d2;        i2 = c2 ? ni2 : i2;
    d1 = c1 ? val : d1;        i1 = c1 ? idx : i1;
}

// Kernel 1: three_nn via V_WMMA_F32_16X16X4_F32.
// One wave handles 16 query points; 256 WMMA tiles scan all M=4096 known pts.
// A-row m = (-2kx, -2ky, -2kz, ||k||^2), B-col n = (ux, uy, uz, 1)
//   => D[m][n] = ||k_m||^2 - 2 k_m . u_n   (||u||^2 added to survivors only).
__global__ __launch_bounds__(256) void three_nn_wmma_kernel(
    const float* __restrict__ unknown,   // [B,N,3]
    const float* __restrict__ known,     // [B,M,3]
    float* __restrict__ wsW,             // [B,N,3] weights
    int*   __restrict__ wsI)             // [B,N,3] indices
{
    __shared__ float4 skn[PM];  // 64 KB: (-2x,-2y,-2z,||k||^2) per known point

    const int t    = threadIdx.x;
    const int wave = t >> 5;
    const int lane = t & 31;

    const int tile    = blockIdx.x * 8 + wave;   // global 16-query tile
    const int tilesPB = PN / 16;                 // 1024, divisible by 8
    const int b       = (blockIdx.x * 8) / tilesPB;  // same b for whole block
    const int n0      = (tile % tilesPB) * 16;

    // Stage pre-scaled known points for this batch into LDS (uniform loop).
    for (int m = t; m < PM; m += 256) {
        const float* kp = known + ((size_t)b * PM + m) * 3;
        float kx = kp[0], ky = kp[1], kz = kp[2];
        skn[m] = make_float4(-2.0f * kx, -2.0f * ky, -2.0f * kz,
                             kx * kx + ky * ky + kz * kz);
    }
    __syncthreads();

    const int  col = lane & 15;
    const bool hi  = lane >= 16;
    const int  n   = n0 + col;

    const float* up = unknown + ((size_t)b * PN + n) * 3;
    float ux = up[0], uy = up[1], uz = up[2];
    float unorm = ux * ux + uy * uy + uz * uz;

    // B-matrix (4x16): lanes 0-15 carry K=0,1; lanes 16-31 carry K=2,3.
    v2f bm;
    bm[0] = hi ? uz : ux;
    bm[1] = hi ? 1.0f : uy;

    float d1 = 3.0e38f, d2 = 3.0e38f, d3 = 3.0e38f;
    int   i1 = 0, i2 = 0, i3 = 0;
    const int rowoff = hi ? 8 : 0;

    for (int m0 = 0; m0 < PM; m0 += 16) {
        float4 s = skn[m0 + col];
        // A-matrix (16x4): lanes 0-15 carry K=0,1; lanes 16-31 carry K=2,3.
        v2f a;
        a[0] = hi ? s.z : s.x;
        a[1] = hi ? s.w : s.y;
        v8f c = {};
        v8f dd = __builtin_amdgcn_wmma_f32_16x16x4_f32(
            false, a, false, bm, (short)0, c, false, false);

        // Per-lane rows: m = m0 + v + rowoff, column n fixed per lane.
        float gmin = fminf(fminf(fminf(dd[0], dd[1]), fminf(dd[2], dd[3])),
                           fminf(fminf(dd[4], dd[5]), fminf(dd[6], dd[7])));
        if (gmin < d3) {
#pragma unroll
            for (int v = 0; v < 8; ++v)
                insert3(dd[v], m0 + v + rowoff, d1, d2, d3, i1, i2, i3);
        }
    }

    // Lane pair (l, l+16) covers the same query: merge their top-3 sets.
    float od1 = __shfl_xor(d1, 16, 32);
    float od2 = __shfl_xor(d2, 16, 32);
    float od3 = __shfl_xor(d3, 16, 32);
    int   oi1 = __shfl_xor(i1, 16, 32);
    int   oi2 = __shfl_xor(i2, 16, 32);
    int   oi3 = __shfl_xor(i3, 16, 32);
    insert3(od1, oi1, d1, d2, d3, i1, i2, i3);
    insert3(od2, oi2, d1, d2, d3, i1, i2, i3);
    insert3(od3, oi3, d1, d2, d3, i1, i2, i3);

    if (!hi) {
        float e1 = d1 + unorm, e2 = d2 + unorm, e3 = d3 + unorm;
        float r1 = 1.0f / (e1 + EPSI);
        float r2 = 1.0f / (e2 + EPSI);
        float r3 = 1.0f / (e3 + EPSI);
        float inv = 1.0f / (r1 + r2 + r3);
        size_t base = ((size_t)b * PN + n) * 3;
        wsW[base + 0] = r1 * inv;
        wsW[base + 1] = r2 * inv;
        wsW[base + 2] = r3 * inv;
        wsI[base + 0] = i1;
        wsI[base + 1] = i2;
        wsI[base + 2] = i3;
    }
}

// Kernel 2: gather + weighted blend. Block = 64 queries x 4 channels/iter;
// weights staged in LDS; output writes fully coalesced; 16KB feature rows
// stay L2-hot (feats total = 32MB << 192MB L2).
__global__ __launch_bounds__(256) void three_interp_kernel(
    const float* __restrict__ feats,   // [B,C,M]
    const float* __restrict__ wsW,
    const int*   __restrict__ wsI,
    float* __restrict__ out)           // [B,C,N]
{
    __shared__ float sw[64 * 3];
    __shared__ int   si[64 * 3];

    const int b  = blockIdx.y;
    const int n0 = blockIdx.x * 64;
    const int t  = threadIdx.x;

    if (t < 192) {
        size_t base = ((size_t)b * PN + n0) * 3;
        sw[t] = wsW[base + t];
        si[t] = wsI[base + t];
    }
    __syncthreads();

    const int nl = t & 63;
    const int cq = t >> 6;  // 0..3
    float w1 = sw[nl * 3 + 0], w2 = sw[nl * 3 + 1], w3 = sw[nl * 3 + 2];
    int   j1 = si[nl * 3 + 0], j2 = si[nl * 3 + 1], j3 = si[nl * 3 + 2];

    const float* fb = feats + (size_t)b * PC * PM;
    float*       ob = out + (size_t)b * PC * PN + n0 + nl;

    for (int c = cq; c < PC; c += 4) {
        const float* fr = fb + (size_t)c * PM;
        float r = w1 * fr[j1] + w2 * fr[j2] + w3 * fr[j3];
        ob[(size_t)c * PN] = r;
    }
}

extern "C" void kernel_launch(void* const* d_in, const int* in_sizes, int n_in,
                              void* d_out, int out_size, void* d_ws, size_t ws_size,
                              hipStream_t stream) {
    const float* unknown = (const float*)d_in[0];  // [8,16384,3]
    const float* known   = (const float*)d_in[1];  // [8,4096,3]
    const float* feats   = (const float*)d_in[2];  // [8,256,4096]
    float*       out     = (float*)d_out;          // [8,256,16384]

    float* wsW = (float*)d_ws;                       // B*N*3 floats
    int*   wsI = (int*)((float*)d_ws + 3 * PB * PN); // B*N*3 ints  (3.1 MB total)

    // 8192 query tiles of 16 points; 8 waves per block.
    three_nn_wmma_kernel<<<dim3((PB * (PN / 16)) / 8), dim3(256), 0, stream>>>(
        unknown, known, wsW, wsI);

    three_interp_kernel<<<dim3(PN / 64, PB), dim3(256), 0, stream>>>(
        feats, wsW, wsI, out);

    (void)in_sizes; (void)n_in; (void)out_size; (void)ws_size;
}